// CrossLayer_8598524526890
// MI455X (gfx1250) — compile-verified
//
#include <hip/hip_runtime.h>

typedef float v2f __attribute__((ext_vector_type(2)));
typedef float v8f __attribute__((ext_vector_type(8)));

#define DDIM 512
#define FDIM 64
#define ODIM 64
#define LDSP 516   // LDS row pitch in floats: 516 % 64 == 4 -> conflict-free A reads
#define MROWS 32   // batch rows per workgroup (2 M-tiles sharing each B fragment)

// ---------------------------------------------------------------------------
// Kernel 1: Wm[i*512+j] = (j>i) ? sum_o kernel[i,fid[j],o]*kernel[j,fid[i],o] : 0
// ---------------------------------------------------------------------------
__global__ __launch_bounds__(256) void build_w_kernel(
    const float* __restrict__ kern, const int* __restrict__ fid,
    float* __restrict__ W) {
  const int t = blockIdx.x * 256 + threadIdx.x;   // 0 .. 512*512-1
  const int i = t >> 9;
  const int j = t & 511;
  float acc = 0.0f;
  if (j > i) {
    const float* ki = kern + (size_t)i * (FDIM * ODIM) + (size_t)fid[j] * ODIM;
    const float* kj = kern + (size_t)j * (FDIM * ODIM) + (size_t)fid[i] * ODIM;
#pragma unroll
    for (int o = 0; o < ODIM; o += 4) {
      const float4 a = *(const float4*)(ki + o);
      const float4 b = *(const float4*)(kj + o);
      acc += a.x * b.x + a.y * b.y + a.z * b.z + a.w * b.w;
    }
  }
  W[t] = acc;
}

// ---------------------------------------------------------------------------
// Kernel 2: out[b] = sum_j (X @ Wm)[b,j] * X[b,j]
// One workgroup per 32 batch rows (two 16-row M-tiles). 8 waves; wave w
// handles column tiles w, w+8, w+16, w+24. Each K-chunk's B fragment (Wm)
// is reused by both M-tiles -> halves global-load issue per WMMA.
// ---------------------------------------------------------------------------
__global__ __launch_bounds__(256) void quad_form_kernel(
    const float* __restrict__ X, const float* __restrict__ W,
    float* __restrict__ out) {
  __shared__ float ldsX[MROWS * LDSP];   // 32 rows of X, padded pitch (~66 KB)
  __shared__ float partial[8][MROWS];    // per-wave row partials

  const int tid   = threadIdx.x;
  const int wave  = tid >> 5;
  const int lane  = tid & 31;
  const int l15   = lane & 15;
  const int half  = lane >> 4;           // 0: lanes 0-15, 1: lanes 16-31
  const int mBase = blockIdx.x * MROWS;

  // Cooperative load of the 32x512 X tile into LDS (float4, padded rows).
  {
    const float4* src = (const float4*)(X + (size_t)mBase * DDIM);
#pragma unroll
    for (int it = 0; it < 16; ++it) {
      const int idx = tid + it * 256;    // 0..4095 float4s
      const int row = idx >> 7;          // 128 float4 per row
      const int col = idx & 127;
      *(float4*)(ldsX + row * LDSP + col * 4) = src[row * 128 + col];
    }
  }
  __syncthreads();

  float rowSum0[8], rowSum1[8];
#pragma unroll
  for (int r = 0; r < 8; ++r) { rowSum0[r] = 0.0f; rowSum1[r] = 0.0f; }

  // Constant trip count -> scalar loop, EXEC stays all-ones around WMMA.
#pragma unroll 1
  for (int t = 0; t < 4; ++t) {
    const int jBase = (wave + t * 8) * 16;
    v8f c0 = {0.f, 0.f, 0.f, 0.f, 0.f, 0.f, 0.f, 0.f};
    v8f c1 = {0.f, 0.f, 0.f, 0.f, 0.f, 0.f, 0.f, 0.f};
    const float* wcol = W + jBase + l15;       // column jBase+l15 of Wm

#pragma unroll 4
    for (int k0 = 0; k0 < DDIM; k0 += 4) {
      const int kA = k0 + half * 2;
      v2f a0, a1, b;
      // B fragment (4x16): lanes 0-15 rows k0,k0+1; lanes 16-31 rows k0+2,k0+3
      b.x = wcol[(size_t)kA * DDIM];
      b.y = wcol[(size_t)(kA + 1) * DDIM];
      // A fragments for both M-tiles (16x4 each), from LDS
      a0.x = ldsX[l15 * LDSP + kA];
      a0.y = ldsX[l15 * LDSP + kA + 1];
      a1.x = ldsX[(16 + l15) * LDSP + kA];
      a1.y = ldsX[(16 + l15) * LDSP + kA + 1];
      c0 = __builtin_amdgcn_wmma_f32_16x16x4_f32(
          false, a0, false, b, (short)0, c0, false, false);
      c1 = __builtin_amdgcn_wmma_f32_16x16x4_f32(
          false, a1, false, b, (short)0, c1, false, false);
    }

    // Fuse the row dot: Z[m, jBase+l15] * X[m, jBase+l15]
#pragma unroll
    for (int r = 0; r < 8; ++r) {
      const int m = r + half * 8;        // C/D layout: VGPR r -> row r (+8 hi half)
      rowSum0[r] += c0[r] * ldsX[m * LDSP + jBase + l15];
      rowSum1[r] += c1[r] * ldsX[(16 + m) * LDSP + jBase + l15];
    }
  }

  // Deterministic xor reduction across the 16 lanes of each half.
#pragma unroll
  for (int r = 0; r < 8; ++r) {
    float v0 = rowSum0[r];
    v0 += __shfl_xor(v0, 1, 32);
    v0 += __shfl_xor(v0, 2, 32);
    v0 += __shfl_xor(v0, 4, 32);
    v0 += __shfl_xor(v0, 8, 32);
    rowSum0[r] = v0;
    float v1 = rowSum1[r];
    v1 += __shfl_xor(v1, 1, 32);
    v1 += __shfl_xor(v1, 2, 32);
    v1 += __shfl_xor(v1, 4, 32);
    v1 += __shfl_xor(v1, 8, 32);
    rowSum1[r] = v1;
  }
  if (l15 == 0) {                        // lane 0 -> rows r / 16+r, lane 16 -> +8
#pragma unroll
    for (int r = 0; r < 8; ++r) {
      partial[wave][r + half * 8]      = rowSum0[r];
      partial[wave][16 + r + half * 8] = rowSum1[r];
    }
  }
  __syncthreads();

  // Fixed-order cross-wave sum (deterministic), then store.
  if (tid < MROWS) {
    float s = 0.0f;
#pragma unroll
    for (int w = 0; w < 8; ++w) s += partial[w][tid];
    out[mBase + tid] = s;
  }
}

// ---------------------------------------------------------------------------
extern "C" void kernel_launch(void* const* d_in, const int* in_sizes, int n_in,
                              void* d_out, int out_size, void* d_ws, size_t ws_size,
                              hipStream_t stream) {
  const float* x    = (const float*)d_in[0];   // (4096, 512) f32
  const float* kern = (const float*)d_in[1];   // (512, 64, 64) f32
  const int*   fid  = (const int*)d_in[2];     // (512,) i32
  float* out = (float*)d_out;                  // (4096,) f32
  float* W   = (float*)d_ws;                   // 512*512*4 = 1 MB scratch

  build_w_kernel<<<(DDIM * DDIM) / 256, 256, 0, stream>>>(kern, fid, W);
  quad_form_kernel<<<4096 / MROWS, 256, 0, stream>>>(x, W, out);
}